// PrototypicalLoss_59596966199581
// MI455X (gfx1250) — compile-verified
//
#include <hip/hip_runtime.h>

#define NCLS   20
#define NSUP   100
#define NQ     300
#define DDIM   640
#define NB     256
#define PS     644                     // padded proto row stride (floats): 644%64=4 -> conflict-free B frags
#define NTILES ((NQ + 15) / 16)        // 19 query M-tiles
#define QPAD   (NTILES * 16)           // 304

typedef __attribute__((ext_vector_type(2))) float v2f;
typedef __attribute__((ext_vector_type(8))) float v8f;

__global__ void zero_out_kernel(float* out) { out[0] = 0.0f; }

__global__ __launch_bounds__(128)
void proto_loss_kernel(const float* __restrict__ sup,
                       const float* __restrict__ qry,
                       const int*   __restrict__ tgt_s,
                       const int*   __restrict__ tgt_q,
                       float*       __restrict__ out)
{
    __shared__ float s_proto[NCLS * PS];     // 51,520 B  prototypes [20][644]
    __shared__ float s_zero[PS];             //  2,576 B  zero row for padded classes 20..31
    __shared__ float s_dist[4][16][32];      //  8,192 B  per-wave distance tile
    __shared__ float s_qsq[QPAD];            //  1,216 B  ||q||^2 (padded rows = 0)
    __shared__ float s_psq[32];              //    128 B  ||p||^2 (cols 20..31 = 0)
    __shared__ float s_inv[NCLS];            //     80 B  1/count per class
                                             // total ~63.7 KB

    const int b    = blockIdx.x;
    const int tid  = threadIdx.x;
    const int lane = tid & 31;
    const int wid  = tid >> 5;

    const float* supB = sup + (size_t)b * NSUP * DDIM;
    const float* qryB = qry + (size_t)b * NQ   * DDIM;
    const int*   tsB  = tgt_s + (size_t)b * NSUP;
    const int*   tqB  = tgt_q + (size_t)b * NQ;

    // ---------- Phase 1a: class counts (reciprocals) + LDS init ----------
    if (tid < NCLS) {
        int cnt = 0;
        for (int s = 0; s < NSUP; ++s) cnt += (tsB[s] == tid) ? 1 : 0;
        s_inv[tid] = (cnt > 0) ? (1.0f / (float)cnt) : 0.0f;
    }
    for (int i = tid; i < NCLS * PS; i += 128) s_proto[i] = 0.0f;
    for (int i = tid; i < PS; i += 128)        s_zero[i]  = 0.0f;
    if (tid < 32) s_psq[tid] = 0.0f;
    __syncthreads();

    // ---------- Phase 1b: prototype accumulation (threads own disjoint d-columns) ----------
    for (int s = 0; s < NSUP; ++s) {
        const int c = tsB[s];
        const float* row = supB + (size_t)s * DDIM;
        for (int d = tid; d < DDIM; d += 128)
            s_proto[c * PS + d] += row[d];
    }
    __syncthreads();
    for (int i = tid; i < NCLS * DDIM; i += 128) {
        const int c = i / DDIM, d = i - c * DDIM;
        s_proto[c * PS + d] *= s_inv[c];
    }
    __syncthreads();

    // ---------- Phase 1c/1d: ||p||^2 and ||q||^2 ----------
    if (tid < NCLS) {
        float acc = 0.0f;
        for (int d = 0; d < DDIM; ++d) {
            const float v = s_proto[tid * PS + d];
            acc += v * v;
        }
        s_psq[tid] = acc;
    }
    for (int q = tid; q < QPAD; q += 128) {
        float acc = 0.0f;
        if (q < NQ) {
            const float* row = qryB + (size_t)q * DDIM;
            for (int d = 0; d < DDIM; ++d) acc += row[d] * row[d];
        }
        s_qsq[q] = acc;
    }
    __syncthreads();

    // ---------- Phase 2: WMMA cross-terms + fused log-softmax NLL ----------
    const int n     = lane & 15;             // B-frag / C-frag column
    const int khalf = (lane >> 4) << 1;      // 0 or 2 (K sub-offset of this lane half)
    const int mh    = (lane >> 4) << 3;      // C-frag row offset: 0 or 8

    // Per-lane B base pointers, selected ONCE (invalid columns -> shared zero row,
    // broadcast read, no exec-mask juggling in the K loop).
    const float* p0 = &s_proto[n * PS];                                    // class n
    const float* p1 = (n + 16 < NCLS) ? &s_proto[(n + 16) * PS] : s_zero;  // class n+16 or zeros

    float local_sum = 0.0f;

    for (int t = wid; t < NTILES; t += 4) {
        const int m0 = t * 16;
        int aRow = m0 + (lane & 15);
        if (aRow > NQ - 1) aRow = NQ - 1;                 // clamp; masked in reduction
        const float* qrow = qryB + (size_t)aRow * DDIM;

        v8f acc0 = {0,0,0,0,0,0,0,0};
        v8f acc1 = {0,0,0,0,0,0,0,0};

        #pragma unroll 8
        for (int k = 0; k < DDIM; k += 4) {
            v2f a;  a.x  = qrow[k + khalf]; a.y  = qrow[k + khalf + 1];
            v2f b0; b0.x = p0[k + khalf];   b0.y = p0[k + khalf + 1];
            v2f b1; b1.x = p1[k + khalf];   b1.y = p1[k + khalf + 1];
            acc0 = __builtin_amdgcn_wmma_f32_16x16x4_f32(false, a, false, b0,
                                                         (short)0, acc0, false, false);
            acc1 = __builtin_amdgcn_wmma_f32_16x16x4_f32(false, a, false, b1,
                                                         (short)0, acc1, false, false);
        }

        // dists = ||q||^2 + ||p||^2 - 2 q.p  -> LDS tile
        #pragma unroll
        for (int v = 0; v < 8; ++v) {
            const int m  = v + mh;
            const float qs = s_qsq[m0 + m];
            s_dist[wid][m][n]      = qs + s_psq[n]      - 2.0f * acc0[v];
            s_dist[wid][m][n + 16] = qs + s_psq[n + 16] - 2.0f * acc1[v];
        }

        // lanes 0..15 each reduce one query row (same-wave LDS dep, no barrier needed)
        if (lane < 16) {
            const int q = m0 + lane;
            if (q < NQ) {
                const float* row = s_dist[wid][lane];
                float mx = -row[0];
                for (int c = 1; c < NCLS; ++c) { const float l = -row[c]; mx = (l > mx) ? l : mx; }
                float ssum = 0.0f;
                for (int c = 0; c < NCLS; ++c) ssum += __expf(-row[c] - mx);
                const int tc = tqB[q];
                local_sum += mx + __logf(ssum) + row[tc];   // = -log_softmax(-d)[tc]
            }
        }
    }

    // wave reduction, then one atomic per wave
    for (int off = 16; off > 0; off >>= 1)
        local_sum += __shfl_xor(local_sum, off, 32);
    if (lane == 0)
        atomicAdd(out, local_sum * (1.0f / ((float)NB * (float)NQ)));
}

extern "C" void kernel_launch(void* const* d_in, const int* in_sizes, int n_in,
                              void* d_out, int out_size, void* d_ws, size_t ws_size,
                              hipStream_t stream)
{
    (void)in_sizes; (void)n_in; (void)d_ws; (void)ws_size; (void)out_size;
    const float* sup = (const float*)d_in[0];
    const float* qry = (const float*)d_in[1];
    const int*   ts  = (const int*)d_in[2];
    const int*   tq  = (const int*)d_in[3];
    float* out = (float*)d_out;

    zero_out_kernel<<<1, 1, 0, stream>>>(out);
    proto_loss_kernel<<<NB, 128, 0, stream>>>(sup, qry, ts, tq, out);
}